// PrediNet_43087111913961
// MI455X (gfx1250) — compile-verified
//
#include <hip/hip_runtime.h>
#include <hip/hip_bf16.h>

typedef __attribute__((ext_vector_type(2))) float v2f;
typedef __attribute__((ext_vector_type(8))) float v8f;

#define BATCH 512
#define NOBJ 64
#define DT 128
#define FLATK 8192          // NOBJ * DT
#define NQ 512              // N_HEADS * KEY_DIM
#define NHEADS 32
#define KD 16
#define NREL 16
#define OUTW 576            // NHEADS * (NREL + 2)

// ---------------------------------------------------------------------------
// Kernel 1: x = concat(obj_seq, tag)  -> X [B, NOBJ, DT]  (row-major, flat)
// ---------------------------------------------------------------------------
__global__ __launch_bounds__(256) void build_x(const float* __restrict__ obj,
                                               float* __restrict__ X) {
    int i = blockIdx.x * 256 + threadIdx.x;        // flat over B*NOBJ*DT
    if (i >= BATCH * NOBJ * DT) return;
    int d  = i & (DT - 1);
    int bn = i >> 7;                               // b*NOBJ + n
    int n  = bn & (NOBJ - 1);
    X[i] = (d < DT - 1) ? obj[bn * (DT - 1) + d] : (float)n;
}

// ---------------------------------------------------------------------------
// Kernel 2: Q = X[512,8192] @ Wq[8192,512], fp32 WMMA 16x16x4.
// One wave (32 threads) computes a 32x32 output tile (2x2 WMMA subtiles).
// grid = (512/32, 512/32, 2)  z selects W_q1 / W_q2.
//
// VGPR layouts per CDNA5 ISA:
//   A 16x4 f32 : lanes 0-15 -> M=lane, v0=K+0, v1=K+1 ; lanes 16-31 -> K+2,K+3
//   B 4x16 f32 : lanes 0-15 -> N=lane, v0=K+0, v1=K+1 ; lanes 16-31 -> K+2,K+3
//   C/D 16x16  : VGPR i, lanes 0-15 -> M=i, N=lane ; lanes 16-31 -> M=8+i
// ---------------------------------------------------------------------------
__global__ __launch_bounds__(32) void gemm_q(const float* __restrict__ X,
                                             const float* __restrict__ Wq1,
                                             const float* __restrict__ Wq2,
                                             float* __restrict__ Q1,
                                             float* __restrict__ Q2) {
    const int lane = threadIdx.x;
    const int l16  = lane & 15;
    const int hi   = lane >> 4;                    // 0 or 1
    const int m0   = blockIdx.x * 32;              // batch-row tile base
    const int n0   = blockIdx.y * 32;              // output-col tile base
    const float* __restrict__ W = blockIdx.z ? Wq2 : Wq1;
    float* __restrict__ Q       = blockIdx.z ? Q2  : Q1;

    v8f acc00 = {}, acc01 = {}, acc10 = {}, acc11 = {};

    const float* arow0 = X + (size_t)(m0 + l16) * FLATK + 2 * hi;
    const float* arow1 = X + (size_t)(m0 + 16 + l16) * FLATK + 2 * hi;

    for (int k = 0; k < FLATK; k += 4) {
        // A fragments (contiguous K pair -> single b64 load each)
        v2f a0 = *(const v2f*)(arow0 + k);
        v2f a1 = *(const v2f*)(arow1 + k);
        // B fragments (two rows of W, stride NQ)
        const float* wk = W + (size_t)(k + 2 * hi) * NQ + n0 + l16;
        v2f b0, b1;
        b0.x = wk[0];        b0.y = wk[NQ];
        b1.x = wk[16];       b1.y = wk[NQ + 16];

        acc00 = __builtin_amdgcn_wmma_f32_16x16x4_f32(false, a0, false, b0,
                                                      (short)0, acc00, false, false);
        acc01 = __builtin_amdgcn_wmma_f32_16x16x4_f32(false, a0, false, b1,
                                                      (short)0, acc01, false, false);
        acc10 = __builtin_amdgcn_wmma_f32_16x16x4_f32(false, a1, false, b0,
                                                      (short)0, acc10, false, false);
        acc11 = __builtin_amdgcn_wmma_f32_16x16x4_f32(false, a1, false, b1,
                                                      (short)0, acc11, false, false);
    }

    // Store: VGPR i holds row (8*hi + i) of the 16x16 subtile, col = l16.
    #pragma unroll
    for (int i = 0; i < 8; ++i) {
        int r0 = m0 + 8 * hi + i;
        int r1 = r0 + 16;
        int c0 = n0 + l16;
        Q[(size_t)r0 * NQ + c0]      = acc00[i];
        Q[(size_t)r0 * NQ + c0 + 16] = acc01[i];
        Q[(size_t)r1 * NQ + c0]      = acc10[i];
        Q[(size_t)r1 * NQ + c0 + 16] = acc11[i];
    }
}

// ---------------------------------------------------------------------------
// Kernel 3: fused attention + relation head, one workgroup per batch element.
// K = x@W_k ; scores = K@Q^T ; softmax over objects ; E = A^T@x ;
// D = (E1-E2)@W_s ; append tag channels of E1,E2.
// q=0,1 handled sequentially to keep LDS under 64 KB.
// ---------------------------------------------------------------------------
__global__ __launch_bounds__(256) void attend_kernel(const float* __restrict__ X,
                                                     const float* __restrict__ Q1g,
                                                     const float* __restrict__ Q2g,
                                                     const float* __restrict__ Wk,
                                                     const float* __restrict__ Ws,
                                                     float* __restrict__ out) {
    __shared__ float xs[NOBJ * DT];        // 32 KB
    __shared__ float Ks[NOBJ * KD];        //  4 KB
    __shared__ float Qs[NHEADS * KD];      //  2 KB
    __shared__ float As[NOBJ * NHEADS];    //  8 KB
    __shared__ float Es[NHEADS * DT];      // 16 KB   (total 62 KB)

    const int b = blockIdx.x;
    const int t = threadIdx.x;

    // stage x_b into LDS (float4 vectorized)
    {
        const float4* src = (const float4*)(X + (size_t)b * FLATK);
        float4* dst = (float4*)xs;
        for (int i = t; i < FLATK / 4; i += 256) dst[i] = src[i];
    }
    __syncthreads();

    // K = x @ W_k   (64x16)
    for (int i = t; i < NOBJ * KD; i += 256) {
        int n = i >> 4, kk = i & 15;
        float s = 0.f;
        const float* xr = xs + n * DT;
        #pragma unroll 4
        for (int d = 0; d < DT; ++d) s += xr[d] * Wk[d * KD + kk];
        Ks[i] = s;
    }

    float dacc0 = 0.f, dacc1 = 0.f;        // two D outputs per thread (512 total)
    const int h0 = t >> 4,        r0 = t & 15;
    const int h1 = (t + 256) >> 4, r1 = t & 15;

    for (int q = 0; q < 2; ++q) {
        const float* __restrict__ Qg = q ? Q2g : Q1g;
        for (int i = t; i < NHEADS * KD; i += 256) Qs[i] = Qg[(size_t)b * NQ + i];
        __syncthreads();

        // scores[n][h] = dot16(K[n], Q[h])
        for (int i = t; i < NOBJ * NHEADS; i += 256) {
            int n = i >> 5, h = i & 31;
            float s = 0.f;
            const float* kr = Ks + n * KD;
            const float* qr = Qs + h * KD;
            #pragma unroll
            for (int kk = 0; kk < KD; ++kk) s += kr[kk] * qr[kk];
            As[i] = s;
        }
        __syncthreads();

        // softmax over n for each head column
        if (t < NHEADS) {
            float mx = -1e30f;
            for (int n = 0; n < NOBJ; ++n) mx = fmaxf(mx, As[n * NHEADS + t]);
            float sum = 0.f;
            for (int n = 0; n < NOBJ; ++n) {
                float e = expf(As[n * NHEADS + t] - mx);
                As[n * NHEADS + t] = e;
                sum += e;
            }
            float inv = 1.f / sum;
            for (int n = 0; n < NOBJ; ++n) As[n * NHEADS + t] *= inv;
        }
        __syncthreads();

        // E[h][d] = sum_n A[n][h] * x[n][d]   (32x128)
        for (int i = t; i < NHEADS * DT; i += 256) {
            int h = i >> 7, d = i & 127;
            float s = 0.f;
            #pragma unroll 4
            for (int n = 0; n < NOBJ; ++n) s += As[n * NHEADS + h] * xs[n * DT + d];
            Es[i] = s;
        }
        __syncthreads();

        // D partials: +E1@Ws, -E2@Ws  ; tag channels
        {
            float s0 = 0.f, s1 = 0.f;
            const float* e0 = Es + h0 * DT;
            const float* e1 = Es + h1 * DT;
            #pragma unroll 4
            for (int d = 0; d < DT; ++d) {
                float w0 = Ws[d * NREL + r0];
                s0 += e0[d] * w0;
                s1 += e1[d] * w0;   // r1 == r0
            }
            if (q == 0) { dacc0 += s0; dacc1 += s1; }
            else        { dacc0 -= s0; dacc1 -= s1; }
        }
        if (t < NHEADS)
            out[(size_t)b * OUTW + t * (NREL + 2) + NREL + q] = Es[t * DT + DT - 1];
        __syncthreads();
    }

    out[(size_t)b * OUTW + h0 * (NREL + 2) + r0] = dacc0;
    out[(size_t)b * OUTW + h1 * (NREL + 2) + r1] = dacc1;
}

// ---------------------------------------------------------------------------
extern "C" void kernel_launch(void* const* d_in, const int* in_sizes, int n_in,
                              void* d_out, int out_size, void* d_ws, size_t ws_size,
                              hipStream_t stream) {
    (void)in_sizes; (void)n_in; (void)out_size; (void)ws_size;
    const float* obj = (const float*)d_in[0];
    const float* Wk  = (const float*)d_in[1];
    const float* Wq1 = (const float*)d_in[2];
    const float* Wq2 = (const float*)d_in[3];
    const float* Ws  = (const float*)d_in[4];
    float* out = (float*)d_out;

    float* X  = (float*)d_ws;                        // 512*8192 floats (16.8 MB)
    float* Q1 = X + (size_t)BATCH * FLATK;           // 512*512
    float* Q2 = Q1 + (size_t)BATCH * NQ;             // 512*512

    build_x<<<(BATCH * NOBJ * DT + 255) / 256, 256, 0, stream>>>(obj, X);

    dim3 gg(BATCH / 32, NQ / 32, 2);
    gemm_q<<<gg, 32, 0, stream>>>(X, Wq1, Wq2, Q1, Q2);

    attend_kernel<<<BATCH, 256, 0, stream>>>(X, Q1, Q2, Wk, Ws, out);
}